// GNNCritic_11845519803074
// MI455X (gfx1250) — compile-verified
//
#include <hip/hip_runtime.h>

// Problem constants (match reference setup_inputs)
#define NN 60000      // nodes = B*A
#define CC 256        // channels (= GEMM K)
#define EE 960000     // edges
#define BG 10000      // graphs
#define AA 6          // act dim
#define HH 32         // hidden

typedef float v2f __attribute__((ext_vector_type(2)));
typedef float v8f __attribute__((ext_vector_type(8)));

// ---------------------------------------------------------------------------
// Elementwise helpers
// ---------------------------------------------------------------------------
__global__ void zero_f32(float* __restrict__ p, int n) {
  int i = blockIdx.x * blockDim.x + threadIdx.x;
  if (i < n) p[i] = 0.0f;
}

__global__ void degree_kernel(const int* __restrict__ dst, float* __restrict__ deg, int e) {
  int i = blockIdx.x * blockDim.x + threadIdx.x;
  if (i < e) atomicAdd(&deg[dst[i]], 1.0f);
}

__global__ void dinv_kernel(const float* __restrict__ deg, float* __restrict__ dinv, int n) {
  int i = blockIdx.x * blockDim.x + threadIdx.x;
  if (i < n) dinv[i] = rsqrtf(deg[i] + 1.0f);
}

// ---------------------------------------------------------------------------
// Edge scatter: one wave per edge; lane owns 8 contiguous channels.
// agg[dst] += xw[src] * dinv[src]*dinv[dst]   (L2-resident atomics)
// ---------------------------------------------------------------------------
__global__ void edge_scatter_kernel(const float* __restrict__ xw, const int* __restrict__ src,
                                    const int* __restrict__ dst, const float* __restrict__ dinv,
                                    float* __restrict__ agg) {
  int e = blockIdx.x * (blockDim.x >> 5) + (threadIdx.x >> 5);
  int lane = threadIdx.x & 31;
  int s = src[e];
  int d = dst[e];
  float w = dinv[s] * dinv[d];
  const float4* xs4 = (const float4*)(xw + (size_t)s * CC) + lane * 2;
  float4 v0 = xs4[0];
  float4 v1 = xs4[1];
  float* ad = agg + (size_t)d * CC + lane * 8;
  atomicAdd(ad + 0, v0.x * w);
  atomicAdd(ad + 1, v0.y * w);
  atomicAdd(ad + 2, v0.z * w);
  atomicAdd(ad + 3, v0.w * w);
  atomicAdd(ad + 4, v1.x * w);
  atomicAdd(ad + 5, v1.y * w);
  atomicAdd(ad + 6, v1.z * w);
  atomicAdd(ad + 7, v1.w * w);
}

// ---------------------------------------------------------------------------
// FP32 WMMA GEMM with LDS-staged B:
//   out[rows, NCOLS] (+)= relu?(X)[rows, 256] @ W[256, NCOLS]
//   optionally also writes agg = out*dinv(row)^2 + bias[col] (GCN self-loop init)
//
// Block = 256 threads = 8 waves. Each block owns one 16*CTW-wide column group;
// the W column block is staged once into LDS in the per-lane WMMA B layout:
//   entry e = (kstep*CTW + j)*32 + lane  ->  { W[k][n], W[k+1][n] }
//   with k = kstep*4 + 2*(lane>>4), n = n0 + j*16 + (lane&15)
// Each wave then computes a 16 x (CTW*16) strip with a depth-1 software
// pipeline: A via global_load_b64 (immediate offsets), B via ds_load_b64.
//
// ISA 7.12.2 32-bit layouts (wave32):
//   A 16x4 : lanes 0-15 -> M=lane, v0=K0 v1=K1 ; lanes 16-31 -> M=lane-16, v0=K2 v1=K3
//   B 4x16 : lanes 0-15 -> N=lane, v0=K0 v1=K1 ; lanes 16-31 -> N=lane-16, v0=K2 v1=K3
//   C/D    : VGPR r: lanes 0-15 -> (M=r, N=lane); lanes 16-31 -> (M=r+8, N=lane-16)
// ---------------------------------------------------------------------------
template <int CTW, bool ACC, bool RELUA, bool AGGI>
__global__ __launch_bounds__(256) void wmma_gemm_f32(
    const float* __restrict__ X, const float* __restrict__ W,
    float* __restrict__ out, float* __restrict__ agg,
    const float* __restrict__ dinv, const float* __restrict__ bias,
    int NCOLS, int rowTiles, int colGroups) {
  constexpr int KSTEPS = CC / 4;                       // 64
  __shared__ float ldsb[KSTEPS * CTW * 32 * 2];        // 64KB (CTW=4) / 32KB (CTW=2)

  int tid = threadIdx.x;
  int wv = tid >> 5;
  int lane = tid & 31;
  int l16 = lane & 15;
  int lh  = lane >> 4;
  int rg = blockIdx.x / colGroups;
  int cg = blockIdx.x % colGroups;
  int n0 = cg * (CTW * 16);

  // ---- stage W column block into LDS (all threads, before any exit) ----
  for (int e = tid; e < KSTEPS * CTW * 32; e += 256) {
    int elane = e & 31;
    int ej    = (e >> 5) % CTW;
    int ekb   = e / (32 * CTW);
    int k = ekb * 4 + 2 * (elane >> 4);
    int n = n0 + ej * 16 + (elane & 15);
    ldsb[e * 2 + 0] = W[(size_t)k * NCOLS + n];
    ldsb[e * 2 + 1] = W[(size_t)(k + 1) * NCOLS + n];
  }
  __syncthreads();

  int rt = rg * 8 + wv;
  if (rt >= rowTiles) return;

  // ---- accumulator init ----
  v8f acc[CTW];
#pragma unroll
  for (int j = 0; j < CTW; ++j) acc[j] = (v8f)0.0f;
  if (ACC) {
#pragma unroll
    for (int r = 0; r < 8; ++r) {
      int row = rt * 16 + r + 8 * lh;
#pragma unroll
      for (int j = 0; j < CTW; ++j)
        acc[j][r] = out[(size_t)row * NCOLS + n0 + j * 16 + l16];
    }
  }

  const float* aptr = X + (size_t)(rt * 16 + l16) * CC + 2 * lh;
  const float* bbase = &ldsb[lane * 2];

  auto loadA = [&](int ks) -> v2f {
    v2f a = *(const v2f*)(aptr + ks * 4);
    if (RELUA) { a.x = fmaxf(a.x, 0.0f); a.y = fmaxf(a.y, 0.0f); }
    return a;
  };
  auto loadB = [&](int ks, int j) -> v2f {
    return *(const v2f*)(bbase + (size_t)(ks * CTW + j) * 64);
  };

  // ---- K loop, software-pipelined with prefetch distance 1 ----
  v2f a_cur = loadA(0);
  v2f b_cur[CTW];
#pragma unroll
  for (int j = 0; j < CTW; ++j) b_cur[j] = loadB(0, j);

#pragma unroll 4
  for (int ks = 0; ks < KSTEPS - 1; ++ks) {
    v2f a_nxt = loadA(ks + 1);
    v2f b_nxt[CTW];
#pragma unroll
    for (int j = 0; j < CTW; ++j) b_nxt[j] = loadB(ks + 1, j);
#pragma unroll
    for (int j = 0; j < CTW; ++j)
      acc[j] = __builtin_amdgcn_wmma_f32_16x16x4_f32(false, a_cur, false, b_cur[j],
                                                     (short)0, acc[j], false, false);
    a_cur = a_nxt;
#pragma unroll
    for (int j = 0; j < CTW; ++j) b_cur[j] = b_nxt[j];
  }
#pragma unroll
  for (int j = 0; j < CTW; ++j)
    acc[j] = __builtin_amdgcn_wmma_f32_16x16x4_f32(false, a_cur, false, b_cur[j],
                                                   (short)0, acc[j], false, false);

  // ---- store (+ fused GCN self-loop/bias init of agg) ----
#pragma unroll
  for (int r = 0; r < 8; ++r) {
    int row = rt * 16 + r + 8 * lh;
    float di2 = 0.0f;
    if (AGGI) { float di = dinv[row]; di2 = di * di; }
#pragma unroll
    for (int j = 0; j < CTW; ++j) {
      int n = n0 + j * 16 + l16;
      float v = acc[j][r];
      out[(size_t)row * NCOLS + n] = v;
      if (AGGI) agg[(size_t)row * NCOLS + n] = v * di2 + bias[n];
    }
  }
}

// ---------------------------------------------------------------------------
// Fused head: per graph b (one wave, lane = hidden channel h):
//   x1[a,h] = relu(lin1_acc[b*6+a, h] + action[b,a]*lin1_w[1536,h] + lin1_b[h])
//   x2[a,h] = relu(sum_k x1[a,k]*lin2_w[k,h] + lin2_b[h]);  s[h] = sum_a x2
//   out[b]  = sum_h s[h]*lin3_w[h] + lin3_b[0]
// ---------------------------------------------------------------------------
__global__ void head_kernel(const float* __restrict__ lin1_acc, const float* __restrict__ action,
                            const float* __restrict__ lin1_w, const float* __restrict__ lin1_b,
                            const float* __restrict__ lin2_w, const float* __restrict__ lin2_b,
                            const float* __restrict__ lin3_w, const float* __restrict__ lin3_b,
                            float* __restrict__ out, int nGraphs) {
  __shared__ float ls[8][AA * HH];                     // 8 waves per block
  int wv = threadIdx.x >> 5;
  int h  = threadIdx.x & 31;
  int b  = blockIdx.x * 8 + wv;

  float w1a = lin1_w[1536 * HH + h];                   // action row of lin1
  float b1h = lin1_b[h];
  if (b < nGraphs) {
#pragma unroll
    for (int a = 0; a < AA; ++a) {
      float v = lin1_acc[(size_t)(b * AA + a) * HH + h] + action[b * AA + a] * w1a + b1h;
      ls[wv][a * HH + h] = fmaxf(v, 0.0f);
    }
  }
  __syncthreads();

  if (b < nGraphs) {
    float b2h = lin2_b[h];
    float sumh = 0.0f;
#pragma unroll
    for (int a = 0; a < AA; ++a) {
      float acc = b2h;
#pragma unroll
      for (int k = 0; k < HH; ++k) acc += ls[wv][a * HH + k] * lin2_w[k * HH + h];
      sumh += fmaxf(acc, 0.0f);
    }
    float p = sumh * lin3_w[h];
#pragma unroll
    for (int off = 16; off > 0; off >>= 1) p += __shfl_xor(p, off, 32);
    if (h == 0) out[b] = p + lin3_b[0];
  }
}

// ---------------------------------------------------------------------------
extern "C" void kernel_launch(void* const* d_in, const int* in_sizes, int n_in,
                              void* d_out, int out_size, void* d_ws, size_t ws_size,
                              hipStream_t stream) {
  (void)in_sizes; (void)n_in; (void)out_size; (void)ws_size;

  const float* state  = (const float*)d_in[0];   // [N, C]
  const int*   eidx   = (const int*)  d_in[1];   // [2, E]: src = eidx, dst = eidx+E
  const float* action = (const float*)d_in[2];   // [B, A]
  const float* W1 = (const float*)d_in[3];
  const float* b1 = (const float*)d_in[4];
  const float* W2 = (const float*)d_in[5];
  const float* b2 = (const float*)d_in[6];
  const float* W3 = (const float*)d_in[7];
  const float* b3 = (const float*)d_in[8];
  // d_in[9..12] = W4,b4,W5,b5: unused (reference reuses conv3)
  const float* lin1_w = (const float*)d_in[13];  // [1537, 32]
  const float* lin1_b = (const float*)d_in[14];
  const float* lin2_w = (const float*)d_in[15];  // [32, 32]
  const float* lin2_b = (const float*)d_in[16];
  const float* lin3_w = (const float*)d_in[17];  // [32, 1]
  const float* lin3_b = (const float*)d_in[18];
  float* out = (float*)d_out;                    // [B]

  const int* src = eidx;
  const int* dst = eidx + EE;

  // Workspace carve-up (floats). Total ~48.12M floats = 192.5 MB.
  const size_t NC = (size_t)NN * CC;
  float* ws    = (float*)d_ws;
  float* xw    = ws;                           // [N, C]
  float* aggB[2] = { ws + NC, ws + 2 * NC };   // ping-pong [N, C] (pre-relu)
  float* lacc  = ws + 3 * NC;                  // [N, 32]
  float* deg   = lacc + (size_t)NN * HH;       // [N]
  float* dinv  = deg + NN;                     // [N]

  const int rowTiles  = NN / 16;               // 3750
  const int rowGroups = (rowTiles + 7) / 8;    // 469 (8 waves per block)
  const int gridMain  = rowGroups * 4;         // 4 col groups of 64
  const int gridLin1  = rowGroups * 1;         // 1 col group of 32

  // Degree + normalization
  zero_f32<<<(NN + 255) / 256, 256, 0, stream>>>(deg, NN);
  degree_kernel<<<(EE + 255) / 256, 256, 0, stream>>>(dst, deg, EE);
  dinv_kernel<<<(NN + 255) / 256, 256, 0, stream>>>(deg, dinv, NN);

  const float* Ws[5] = { W1, W2, W3, W3, W3 };
  const float* bs[5] = { b1, b2, b3, b3, b3 };
  const float* x = state;                      // layer 0 input: raw state (no relu)

  for (int l = 0; l < 5; ++l) {
    float* agg = aggB[l & 1];
    // xw = relu?(x) @ W_l ; agg = xw*dinv^2 + b  (fused self-loop + bias init)
    if (l == 0)
      wmma_gemm_f32<4, false, false, true><<<gridMain, 256, 0, stream>>>(
          x, Ws[l], xw, agg, dinv, bs[l], CC, rowTiles, 4);
    else
      wmma_gemm_f32<4, false, true, true><<<gridMain, 256, 0, stream>>>(
          x, Ws[l], xw, agg, dinv, bs[l], CC, rowTiles, 4);
    // agg[dst] += xw[src] * dinv[src]*dinv[dst]
    edge_scatter_kernel<<<EE / 8, 256, 0, stream>>>(xw, src, dst, dinv, agg);
    // lin1_acc (+)= relu(agg) @ lin1_w[256*l : 256*(l+1), :]
    if (l == 0)
      wmma_gemm_f32<2, false, true, false><<<gridLin1, 256, 0, stream>>>(
          agg, lin1_w, lacc, nullptr, nullptr, nullptr, HH, rowTiles, 1);
    else
      wmma_gemm_f32<2, true, true, false><<<gridLin1, 256, 0, stream>>>(
          agg, lin1_w + (size_t)l * CC * HH, lacc, nullptr, nullptr, nullptr, HH, rowTiles, 1);
    x = agg;                                   // next layer applies relu on load
  }

  // lin1_acc += state @ lin1_w[1280:1536, :]   (raw state, no relu)
  wmma_gemm_f32<2, true, false, false><<<gridLin1, 256, 0, stream>>>(
      state, lin1_w + (size_t)5 * CC * HH, lacc, nullptr, nullptr, nullptr, HH, rowTiles, 1);

  // Fused MLP head + readout
  head_kernel<<<(BG + 7) / 8, 256, 0, stream>>>(lacc, action, lin1_w, lin1_b,
                                                lin2_w, lin2_b, lin3_w, lin3_b, out, BG);
}